// VQC_class_43215960932476
// MI455X (gfx1250) — compile-verified
//
#include <hip/hip_runtime.h>
#include <hip/hip_bf16.h>
#include <math.h>
#include <stdint.h>

typedef __attribute__((ext_vector_type(2))) float v2f;
typedef __attribute__((ext_vector_type(8))) float v8f;
typedef __attribute__((ext_vector_type(4))) unsigned int u32x4;
typedef __attribute__((ext_vector_type(8))) int i32x8;
typedef __attribute__((ext_vector_type(4))) int i32x4;

#define NQ      10
#define DIM     1024      // 2^NQ
#define NLAYERS 4
#define NOUT    16
#define BTILE   16        // batch rows per workgroup
#define TPB     512       // 16 waves of 32

// ---------------------------------------------------------------------------
// Prep kernel: builds into d_ws
//   ws[0 .. 16384)         : V table PRE-SWIZZLED into WMMA B-fragment order.
//       slot = s*32 + lane (s = K-step, lane = half*16 + n) holds the float2
//       { V[4s+2*half][n], V[4s+2*half+1][n] }  so each lane's B-frag is one
//       contiguous 8-byte load (global_load_b64, 256B coalesced per wave).
//   ws[16384 .. 16384+320) : per-gate complex 2x2 coefficients (40 gates x 8)
// ---------------------------------------------------------------------------
__global__ void vqc_prep(const float* __restrict__ wts,   // (4,10,3)
                         const float* __restrict__ W,     // (16,10)
                         float* __restrict__ ws) {
  int gid = blockIdx.x * blockDim.x + threadIdx.x;
  if (gid < DIM * NOUT) {
    int f    = gid & 1;
    int slot = gid >> 1;
    int lane = slot & 31;
    int s    = slot >> 5;
    int n    = lane & 15;
    int half = lane >> 4;
    int i    = 4 * s + 2 * half + f;     // basis index (K)
    float v = 0.f;
#pragma unroll
    for (int q = 0; q < NQ; ++q) {
      float w = W[n * NQ + q];
      v += (((i >> (NQ - 1 - q)) & 1) ? -w : w);
    }
    ws[gid] = v;
  } else if (gid < DIM * NOUT + NLAYERS * NQ) {
    int g = gid - DIM * NOUT;                 // g = l*10 + q
    float phi = wts[g * 3 + 0], th = wts[g * 3 + 1], om = wts[g * 3 + 2];
    float sh, ch; sincosf(0.5f * th, &sh, &ch);
    float sp, cp; sincosf(0.5f * (phi + om), &sp, &cp);
    float sm, cm; sincosf(0.5f * (phi - om), &sm, &cm);
    float* gc = ws + DIM * NOUT + g * 8;
    gc[0] =  cp * ch;  gc[1] = -sp * ch;     // a = e^{-i(phi+om)/2} cos
    gc[2] = -cm * sh;  gc[3] = -sm * sh;     // b = -e^{+i(phi-om)/2} sin
    gc[4] =  cm * sh;  gc[5] = -sm * sh;     // c = e^{-i(phi-om)/2} sin
    gc[6] =  cp * ch;  gc[7] =  sp * ch;     // d = e^{+i(phi+om)/2} cos
  }
}

// ---------------------------------------------------------------------------
// Main fused kernel: one workgroup simulates BTILE=16 batch rows with the
// full 16x1024 complex state resident in LDS (128 KB of the 320 KB/WGP).
// X tile arrives via one TDM tensor_load_to_lds (64 KB, 2D tile 1024x16 f32);
// the readout contracts probs (16x1024) against V (1024x16) with f32 WMMA.
// ---------------------------------------------------------------------------
__global__ __launch_bounds__(TPB, 1)
void vqc_main(const float* __restrict__ X,
              const float* __restrict__ Vt,     // swizzled B-frag table
              const float* __restrict__ GC,     // 40 x 8 gate coeffs
              const float* __restrict__ bias,   // 16
              float* __restrict__ out) {        // 4096 x 16
  __shared__ float sre[BTILE * DIM];   // 64 KB
  __shared__ float sim[BTILE * DIM];   // 64 KB
  __shared__ float acc[BTILE * NOUT];  // 16x16 output tile

  const int tid  = threadIdx.x;
  const int wave = tid >> 5;
  const int lane = tid & 31;
  const int b0   = blockIdx.x * BTILE;

  // warm V table into L2/WGP$ early: 512 threads x 128B = whole 64 KB
  __builtin_prefetch(Vt + tid * 32, 0, 0);

  // ---- TDM: DMA the 16x1024 f32 X tile straight into LDS (sre) ------------
  if (wave == 0) {
    uint64_t gaddr  = (uint64_t)(uintptr_t)X + (uint64_t)blockIdx.x * (BTILE * DIM * 4);
    uint32_t ldsoff = (uint32_t)(uintptr_t)(&sre[0]);      // LDS byte offset
    // D# group0: [1:0]=count=1 | lds_addr | global_addr[56:0] | type=2
    u32x4 g0 = { 1u, ldsoff, (uint32_t)gaddr,
                 (uint32_t)((gaddr >> 32) & 0x01FFFFFFu) | 0x80000000u };
    // D# group1: data_size=2(4B); tensor_dim0=1024, tensor_dim1=16;
    //            tile_dim0=1024, tile_dim1=16; tensor_dim0_stride=1024
    i32x8 g1 = { 0x00020000,
                 (int)(DIM   << 16),      // tensor_dim0[15:0]  -> bits 63:48
                 (int)(BTILE << 16),      // tensor_dim1[15:0]  -> bits 95:80
                 (int)(DIM   << 16),      // tile_dim0          -> bits 127:112
                 BTILE,                   // tile_dim1
                 DIM,                     // tensor_dim0_stride
                 0, 0 };
    i32x4 gz4 = { 0, 0, 0, 0 };
    i32x8 gz8 = { 0, 0, 0, 0, 0, 0, 0, 0 };
    __builtin_amdgcn_tensor_load_to_lds(g0, g1, gz4, gz4, gz8, 0);
    __builtin_amdgcn_s_wait_tensorcnt(0);
  }
  __syncthreads();

  // ---- normalize in LDS: wave w owns batch row w --------------------------
  {
    float v[32];
    float ssum = 0.f;
#pragma unroll
    for (int k = 0; k < 32; ++k) {
      v[k] = sre[wave * DIM + lane + 32 * k];
      ssum += v[k] * v[k];
    }
#pragma unroll
    for (int m = 16; m >= 1; m >>= 1)       // wave32 butterfly reduce
      ssum += __shfl_xor(ssum, m, 32);
    float rn = 1.0f / sqrtf(ssum);
#pragma unroll
    for (int k = 0; k < 32; ++k) {
      sre[wave * DIM + lane + 32 * k] = v[k] * rn;
      sim[wave * DIM + lane + 32 * k] = 0.f;
    }
  }
  __syncthreads();

  // ---- circuit: 4 layers of 10 rotations + 10-CNOT ring -------------------
  for (int l = 0; l < NLAYERS; ++l) {
    for (int q = 0; q < NQ; ++q) {
      const float* u = GC + (l * NQ + q) * 8;  // wave-uniform -> s_load
      float ar = u[0], ai = u[1], br = u[2], bi = u[3];
      float cr = u[4], ci = u[5], dr = u[6], di = u[7];
      const int sh = NQ - 1 - q;               // bit position of qubit q
      const int Bm = 1 << sh;                  // pair stride
      for (int pp = tid; pp < BTILE * (DIM / 2); pp += TPB) {
        int bb = pp >> (NQ - 1);
        int p  = pp & (DIM / 2 - 1);
        int i0 = ((p >> sh) << (sh + 1)) | (p & (Bm - 1));
        int base = bb * DIM + i0;
        float s0r = sre[base],      s0i = sim[base];
        float s1r = sre[base + Bm], s1i = sim[base + Bm];
        float n0r = ar * s0r - ai * s0i + br * s1r - bi * s1i;
        float n0i = ar * s0i + ai * s0r + br * s1i + bi * s1r;
        float n1r = cr * s0r - ci * s0i + dr * s1r - di * s1i;
        float n1i = cr * s0i + ci * s0r + dr * s1i + di * s1r;
        sre[base]      = n0r;  sim[base]      = n0i;
        sre[base + Bm] = n1r;  sim[base + Bm] = n1i;
      }
      __syncthreads();
    }
    const int r = l % (NQ - 1) + 1;            // = l+1 for l<9
    for (int q = 0; q < NQ; ++q) {
      const int t  = (q + r) % NQ;
      const int pc = NQ - 1 - q, pt = NQ - 1 - t;
      const int maskC = 1 << pc, maskT = 1 << pt;
      const int plo = (pc < pt) ? pc : pt;
      const int phb = (pc < pt) ? pt : pc;
      // swap amp(i: c=1,t=0) <-> amp(i^maskT): 256 disjoint swaps per row
      for (int ss = tid; ss < BTILE * (DIM / 4); ss += TPB) {
        int bb = ss >> (NQ - 2);
        int k  = ss & (DIM / 4 - 1);
        int i  = ((k >> plo) << (plo + 1)) | (k & ((1 << plo) - 1));
        i      = ((i >> phb) << (phb + 1)) | (i & ((1 << phb) - 1));
        i |= maskC;                            // control=1, target bit is 0
        int j = i ^ maskT;
        int base = bb * DIM;
        float tr = sre[base + i], ti = sim[base + i];
        sre[base + i] = sre[base + j];  sim[base + i] = sim[base + j];
        sre[base + j] = tr;             sim[base + j] = ti;
      }
      __syncthreads();
    }
  }

  // ---- probabilities in place (re <- re^2 + im^2) -------------------------
  for (int j = tid; j < BTILE * DIM; j += TPB) {
    float re = sre[j], im = sim[j];
    sre[j] = re * re + im * im;
  }
  if (tid < BTILE * NOUT) acc[tid] = 0.f;
  __syncthreads();

  // ---- probs(16x1024) @ V(1024x16) via V_WMMA_F32_16X16X4_F32 -------------
  // Wave w owns K-steps [16w, 16w+16). B fragments preloaded (swizzled table,
  // one global_load_b64 each), then a fully unrolled chain of 16 WMMAs fed by
  // ds_load_b64 A fragments. Partial C reduced through ds_add_f32.
  {
    const int m    = lane & 15;
    const int half = lane >> 4;
    const float* vb = Vt + (size_t)(wave * 512 + lane) * 2;  // slot*2 floats
    v2f bf[16];
#pragma unroll
    for (int u = 0; u < 16; ++u)
      bf[u] = *(const v2f*)(vb + u * 64);                    // 32 slots/step
    v8f c8 = {};
#pragma unroll
    for (int u = 0; u < 16; ++u) {
      const int s  = wave * 16 + u;
      const int k0 = 4 * s + 2 * half;
      v2f a;
      // A 16x4 f32 layout: lanes 0-15 -> K=0,1 ; lanes 16-31 -> K=2,3
      a.x = sre[m * DIM + k0];
      a.y = sre[m * DIM + k0 + 1];
      c8 = __builtin_amdgcn_wmma_f32_16x16x4_f32(
              false, a, false, bf[u], (short)0, c8, false, false);
    }
#pragma unroll
    for (int rr = 0; rr < 8; ++rr) {
      // C/D layout: VGPR rr holds M = rr (lanes 0-15) / M = rr+8 (lanes 16-31)
      atomicAdd(&acc[(rr + 8 * half) * NOUT + m], c8[rr]);
    }
  }
  __syncthreads();

  // ---- bias + store 16x16 tile -------------------------------------------
  if (tid < BTILE * NOUT) {
    int o = tid & 15, mm = tid >> 4;
    out[(size_t)(b0 + mm) * NOUT + o] = acc[tid] + bias[o];
  }
}

// ---------------------------------------------------------------------------
extern "C" void kernel_launch(void* const* d_in, const int* in_sizes, int n_in,
                              void* d_out, int out_size, void* d_ws, size_t ws_size,
                              hipStream_t stream) {
  const float* X    = (const float*)d_in[0];   // (4096,1024)
  const float* wts  = (const float*)d_in[1];   // (4,10,3)
  const float* W    = (const float*)d_in[2];   // (16,10)
  const float* bias = (const float*)d_in[3];   // (16,)
  float* out = (float*)d_out;
  float* ws  = (float*)d_ws;                   // needs 16384+320 floats (~66 KB)

  const int prep_elems = DIM * NOUT + NLAYERS * NQ;
  vqc_prep<<<(prep_elems + 255) / 256, 256, 0, stream>>>(wts, W, ws);
  vqc_main<<<4096 / BTILE, TPB, 0, stream>>>(X, ws, ws + DIM * NOUT, bias, out);
}